// QuantizedSelectiveScan_30073361006724
// MI455X (gfx1250) — compile-verified
//
#include <hip/hip_runtime.h>
#include <stdint.h>

// Problem constants (from reference setup_inputs)
#define B_SZ     2
#define L_SEQ    4096
#define D_MODEL  2048
#define D_INNER  2048
#define DT_RANK  128
#define D_STATE  16
#define N_OUT    (DT_RANK + 2*D_STATE)   // 160
#define M_TOT    (B_SZ*L_SEQ)            // 8192
#define TCH      64                      // scan timestep chunk staged in LDS

typedef float v2f __attribute__((ext_vector_type(2)));
typedef float v4f __attribute__((ext_vector_type(4)));
typedef float v8f __attribute__((ext_vector_type(8)));

// ---------- helpers ----------

// symmetric int8 fake-quant: clip(round(v/s), -128, 127) * s
__device__ __forceinline__ float fq8(float v, float s) {
  float q = rintf(v / s);
  q = fminf(fmaxf(q, -128.0f), 127.0f);
  return q * s;
}
__device__ __forceinline__ float mkscale(unsigned mbits) {
  return fmaxf(__uint_as_float(mbits) * (1.0f / 127.0f), 1e-8f);
}
__device__ __forceinline__ float wave_max(float v) {
#pragma unroll
  for (int o = 16; o > 0; o >>= 1) v = fmaxf(v, __shfl_xor(v, o, 32));
  return v;
}
__device__ __forceinline__ float block_absmax(float v, float* red) {
  v = wave_max(v);
  const int wid = threadIdx.x >> 5;
  const int nw  = blockDim.x >> 5;
  if ((threadIdx.x & 31) == 0) red[wid] = v;
  __syncthreads();
  float r = (threadIdx.x < (unsigned)nw) ? red[threadIdx.x] : 0.0f;
  r = wave_max(r);
  if (threadIdx.x == 0) red[0] = r;
  __syncthreads();
  return red[0];
}

// ---------- small utility kernels ----------

__global__ void k_init(unsigned* amax) {
  if (threadIdx.x < 16) amax[threadIdx.x] = 0u;
}

__global__ void k_absmax(const float* __restrict__ p, size_t n, unsigned* slot) {
  __shared__ float red[32];
  float m = 0.0f;
  for (size_t i = (size_t)blockIdx.x * blockDim.x + threadIdx.x; i < n;
       i += (size_t)gridDim.x * blockDim.x)
    m = fmaxf(m, fabsf(p[i]));
  m = block_absmax(m, red);
  if (threadIdx.x == 0) atomicMax(slot, __float_as_uint(m));
}

// per-output-channel fake quant of a [rows x cols] row-major weight; 1 block/row
__global__ void k_quant_rows(const float* __restrict__ W, float* __restrict__ Wq, int cols) {
  __shared__ float red[32];
  const int r = blockIdx.x;
  const float* row = W + (size_t)r * cols;
  float m = 0.0f;
  for (int c = threadIdx.x; c < cols; c += blockDim.x) m = fmaxf(m, fabsf(row[c]));
  m = block_absmax(m, red);
  const float s = fmaxf(m * (1.0f / 127.0f), 1e-8f);
  float* orow = Wq + (size_t)r * cols;
  for (int c = threadIdx.x; c < cols; c += blockDim.x) orow[c] = fq8(row[c], s);
}

// A = -exp(fq_per_channel(A_log));   one thread per channel (16 states each)
__global__ void k_prep_A(const float* __restrict__ A_log, float* __restrict__ A) {
  const int r = blockIdx.x * blockDim.x + threadIdx.x;
  if (r >= D_INNER) return;
  const float* row = A_log + (size_t)r * D_STATE;
  float m = 0.0f;
#pragma unroll
  for (int n = 0; n < D_STATE; ++n) m = fmaxf(m, fabsf(row[n]));
  const float s = fmaxf(m * (1.0f / 127.0f), 1e-8f);
  float* o = A + (size_t)r * D_STATE;
#pragma unroll
  for (int n = 0; n < D_STATE; ++n) o[n] = -expf(fq8(row[n], s));
}

// ---------- fp32 WMMA GEMM (row-panel per wave) ----------
// D(16x16) += A(16x4) x B(4x16) via V_WMMA_F32_16X16X4_F32.
// A fragment: lane l holds A[l&15][2*(l>>4)+j]   (contiguous float2 in K)
// B = W^T:    lane l holds W[n0+(l&15)][2*(l>>4)+j] (contiguous float2 in K)
// One wave computes NT tiles of the same 16-row panel: A fragment loaded once
// per K-chunk and reused by NT WMMAs (B tiles at constant IOFFSETs).

// GEMM1: x_dbl[m][n] = x[m][:] . Wxq[n][:]; prompt added to C cols;
// absmax(Bs)->amax[2], absmax(Cs+prompt)->amax[3]. 10 tiles (N=160) per wave.
__global__ void __launch_bounds__(256) k_gemm1(const float* __restrict__ x,
                                               const float* __restrict__ Wxq,
                                               const float* __restrict__ prompt,
                                               float* __restrict__ xdbl,
                                               unsigned* __restrict__ amax) {
  const int NT = N_OUT / 16;                    // 10
  const int wave = (int)((blockIdx.x * blockDim.x + threadIdx.x) >> 5);
  const int tm = wave;                          // row-panel index [0,512)
  const int lane = threadIdx.x & 31, half = lane >> 4, idx = lane & 15;

  const float* ar = x   + (size_t)(tm * 16 + idx) * D_MODEL + 2 * half;
  const float* br = Wxq + (size_t)idx * D_MODEL + 2 * half;

  v8f acc[NT];
#pragma unroll
  for (int t = 0; t < NT; ++t) acc[t] = (v8f){0.f,0.f,0.f,0.f,0.f,0.f,0.f,0.f};

  for (int k = 0; k < D_MODEL; k += 4) {
    const v2f a = *(const v2f*)(ar + k);
#pragma unroll
    for (int t = 0; t < NT; ++t) {
      const v2f b = *(const v2f*)(br + (size_t)t * 16 * D_MODEL + k);
      acc[t] = __builtin_amdgcn_wmma_f32_16x16x4_f32(false, a, false, b, (short)0,
                                                     acc[t], false, false);
    }
  }

  float lmB = 0.0f, lmC = 0.0f;
#pragma unroll
  for (int t = 0; t < NT; ++t) {
    const int n0 = t * 16, n = n0 + idx;
    const bool isB = (n0 == DT_RANK);
    const bool isC = (n0 == DT_RANK + D_STATE);
#pragma unroll
    for (int r = 0; r < 8; ++r) {
      const int m = tm * 16 + half * 8 + r;     // m = b*L + l
      float v = acc[t][r];
      if (isC) { v += prompt[(size_t)m * D_STATE + idx]; lmC = fmaxf(lmC, fabsf(v)); }
      if (isB) lmB = fmaxf(lmB, fabsf(v));
      xdbl[(size_t)m * N_OUT + n] = v;
    }
  }
  lmB = wave_max(lmB);
  lmC = wave_max(lmC);
  if (lane == 0) {
    atomicMax(&amax[2], __float_as_uint(lmB));
    atomicMax(&amax[3], __float_as_uint(lmC));
  }
}

// GEMM2: dt_raw[m][n] = x_dbl[m][:128] . Wdtq[n][:] + bdt[n]; absmax->amax[1].
// 8 N-tiles per wave (N=2048 -> 16 groups of 8).
__global__ void __launch_bounds__(256) k_gemm2(const float* __restrict__ xdbl,
                                               const float* __restrict__ Wdtq,
                                               const float* __restrict__ bdt,
                                               float* __restrict__ dtr,
                                               unsigned* __restrict__ amax) {
  const int NT = 8;
  const int GR = D_INNER / (16 * NT);           // 16 tile-groups in N
  const int wave = (int)((blockIdx.x * blockDim.x + threadIdx.x) >> 5);
  const int tm = wave / GR;                     // [0,512)
  const int tg = wave % GR;                     // [0,16)
  const int lane = threadIdx.x & 31, half = lane >> 4, idx = lane & 15;

  const float* ar = xdbl + (size_t)(tm * 16 + idx) * N_OUT + 2 * half;
  const float* br = Wdtq + (size_t)(tg * 16 * NT + idx) * DT_RANK + 2 * half;

  v8f acc[NT];
#pragma unroll
  for (int t = 0; t < NT; ++t) acc[t] = (v8f){0.f,0.f,0.f,0.f,0.f,0.f,0.f,0.f};

  for (int k = 0; k < DT_RANK; k += 4) {
    const v2f a = *(const v2f*)(ar + k);
#pragma unroll
    for (int t = 0; t < NT; ++t) {
      const v2f b = *(const v2f*)(br + (size_t)t * 16 * DT_RANK + k);
      acc[t] = __builtin_amdgcn_wmma_f32_16x16x4_f32(false, a, false, b, (short)0,
                                                     acc[t], false, false);
    }
  }

  float lm = 0.0f;
#pragma unroll
  for (int t = 0; t < NT; ++t) {
    const int n = (tg * NT + t) * 16 + idx;
    const float bias = bdt[n];
#pragma unroll
    for (int r = 0; r < 8; ++r) {
      const int m = tm * 16 + half * 8 + r;
      const float v = acc[t][r] + bias;
      lm = fmaxf(lm, fabsf(v));
      dtr[(size_t)m * D_INNER + n] = v;
    }
  }
  lm = wave_max(lm);
  if (lane == 0) atomicMax(&amax[1], __float_as_uint(lm));
}

// ---------- selective scan ----------
// 1 lane = 1 (b,d) channel; sequential over L. B_t/C_t (32 contiguous floats
// per step in x_dbl cols 128..159) are async-staged into LDS, double buffered.
__global__ void __launch_bounds__(256) k_scan(const float* __restrict__ x,
                                              const float* __restrict__ xdbl,
                                              const float* __restrict__ dtr,
                                              const float* __restrict__ Aarr,
                                              const float* __restrict__ Dskip,
                                              const unsigned* __restrict__ amax,
                                              float* __restrict__ out) {
  __shared__ float bc[2][TCH][32];              // [buf][step][0:16)=B, [16:32)=C
  const int tid = threadIdx.x;
  const int b = blockIdx.x >> 3;
  const int d = ((blockIdx.x & 7) << 8) + tid;

  const float sX  = mkscale(amax[0]);
  const float sDT = mkscale(amax[1]);
  const float sB  = mkscale(amax[2]);
  const float sC  = mkscale(amax[3]);
  const float sD  = mkscale(amax[4]);
  const float Dq  = fq8(Dskip[d], sD);

  float Av[D_STATE], h[D_STATE];
#pragma unroll
  for (int n = 0; n < D_STATE; ++n) { Av[n] = Aarr[(size_t)d * D_STATE + n]; h[n] = 0.0f; }

  const float* up    = x    + ((size_t)b * L_SEQ) * D_INNER + d;
  const float* dp    = dtr  + ((size_t)b * L_SEQ) * D_INNER + d;
  float*       op    = out  + ((size_t)b * L_SEQ) * D_INNER + d;
  const float* bcsrc = xdbl + ((size_t)b * L_SEQ) * N_OUT + DT_RANK;

  // async-load one TCH-step chunk of B/C (512 b128 segments / block,
  // 2 async instructions per wave -> ASYNCcnt +2 per chunk per wave)
  auto issue = [&](int t0, int buf) {
#pragma unroll
    for (int i = 0; i < 2; ++i) {
      const int fid = i * 256 + tid;            // [0,512)
      const int step = fid >> 3, q = fid & 7;
      const float* g = bcsrc + (size_t)(t0 + step) * N_OUT + q * 4;
      const unsigned lds = (unsigned)(uintptr_t)&bc[buf][step][q * 4];
      asm volatile("global_load_async_to_lds_b128 %0, %1, off"
                   :: "v"(lds), "v"(g) : "memory");
    }
  };

  issue(0, 0);
  const int NCH = L_SEQ / TCH;
  for (int c = 0; c < NCH; ++c) {
    const int buf = c & 1;
    if (c + 1 < NCH) {
      issue((c + 1) * TCH, buf ^ 1);
      asm volatile("s_wait_asynccnt 2" ::: "memory");  // chunk c done, c+1 in flight
    } else {
      asm volatile("s_wait_asynccnt 0" ::: "memory");
    }
    __syncthreads();
    // fake-quant B/C in place once per chunk (shared by all 256 lanes)
#pragma unroll
    for (int j = 0; j < 8; ++j) {
      const int e = tid * 8 + j;
      const int step = e >> 5, col = e & 31;
      bc[buf][step][col] = fq8(bc[buf][step][col], (col < D_STATE) ? sB : sC);
    }
    __syncthreads();

    for (int t = 0; t < TCH; ++t) {
      const int l = c * TCH + t;
      const float uq  = fq8(up[(size_t)l * D_INNER], sX);
      const float dtq = fq8(dp[(size_t)l * D_INNER], sDT);
      const float dt  = (dtq > 20.0f) ? dtq : log1pf(expf(dtq));  // softplus
      const float du  = dt * uq;

      // vector LDS reads: 8x ds_load_b128 instead of 32x b32
      const v4f* p4 = (const v4f*)&bc[buf][t][0];
      v4f Bv[4], Cv[4];
#pragma unroll
      for (int q = 0; q < 4; ++q) { Bv[q] = p4[q]; Cv[q] = p4[4 + q]; }

      float y = 0.0f;
#pragma unroll
      for (int n = 0; n < D_STATE; ++n) {
        const float dA = expf(dt * Av[n]);
        h[n] = fmaf(dA, h[n], du * Bv[n >> 2][n & 3]);
        y = fmaf(h[n], Cv[n >> 2][n & 3], y);
      }
      op[(size_t)l * D_INNER] = y + uq * Dq;
      if (l + 8 < L_SEQ) {                       // global_prefetch_b8 hints
        __builtin_prefetch(up + (size_t)(l + 8) * D_INNER, 0, 0);
        __builtin_prefetch(dp + (size_t)(l + 8) * D_INNER, 0, 0);
      }
    }
    __syncthreads();  // nobody still reading buf when next issue overwrites it
  }
}

// ---------- host launcher ----------
extern "C" void kernel_launch(void* const* d_in, const int* in_sizes, int n_in,
                              void* d_out, int out_size, void* d_ws, size_t ws_size,
                              hipStream_t stream) {
  const float* x      = (const float*)d_in[0];
  const float* prompt = (const float*)d_in[1];
  const float* Wx     = (const float*)d_in[2];
  const float* Wdt    = (const float*)d_in[3];
  const float* bdt    = (const float*)d_in[4];
  const float* A_log  = (const float*)d_in[5];
  const float* Dskip  = (const float*)d_in[6];
  float* out = (float*)d_out;

  // workspace layout (floats): amax[16] | Wxq | Wdtq | A | x_dbl | dt_raw  (~75 MB)
  float* ws      = (float*)d_ws;
  unsigned* amax = (unsigned*)ws;
  float* Wxq  = ws + 16;
  float* Wdtq = Wxq  + (size_t)N_OUT  * D_MODEL;
  float* A    = Wdtq + (size_t)D_INNER * DT_RANK;
  float* xdbl = A    + (size_t)D_INNER * D_STATE;
  float* dtr  = xdbl + (size_t)M_TOT   * N_OUT;

  k_init<<<1, 32, 0, stream>>>(amax);
  k_absmax<<<1024, 256, 0, stream>>>(x, (size_t)M_TOT * D_MODEL, &amax[0]);
  k_absmax<<<8, 256, 0, stream>>>(Dskip, (size_t)D_INNER, &amax[4]);
  k_quant_rows<<<N_OUT, 256, 0, stream>>>(Wx, Wxq, D_MODEL);
  k_quant_rows<<<D_INNER, 128, 0, stream>>>(Wdt, Wdtq, DT_RANK);
  k_prep_A<<<D_INNER / 256, 256, 0, stream>>>(A_log, A);
  // GEMM1: one wave per 16-row panel (512 waves)
  k_gemm1<<<(M_TOT / 16) / 8, 256, 0, stream>>>(x, Wxq, prompt, xdbl, amax);
  // GEMM2: 512 row-panels x 16 tile-groups = 8192 waves
  k_gemm2<<<(M_TOT / 16) * (D_INNER / 128) / 8, 256, 0, stream>>>(xdbl, Wdtq, bdt, dtr, amax);
  k_scan<<<B_SZ * D_INNER / 256, 256, 0, stream>>>(x, xdbl, dtr, A, Dskip, amax, out);
}